// SparseGATLayer_45827301048840
// MI455X (gfx1250) — compile-verified
//
#include <hip/hip_runtime.h>
#include <hip/hip_bf16.h>

typedef float v2f __attribute__((ext_vector_type(2)));
typedef float v8f __attribute__((ext_vector_type(8)));

#define N_NODES   100000
#define E_EDGES   1000000
#define IN_F      256
#define OUT_F     128
#define NEG_SLOPE 0.2f

// ---------------------------------------------------------------------------
// Kernel 1: h = X @ W^T using V_WMMA_F32_16X16X4_F32.
// Block = 256 threads = 8 waves. Block owns one 16-row M tile x all 128 cols.
// Wave w owns output cols [16w, 16w+16). X tile staged in LDS (row pad +4
// floats -> 4-bank rotation per row, no 16-way conflicts on fragment reads).
// ---------------------------------------------------------------------------
__global__ __launch_bounds__(256)
void gat_gemm_wmma(const float* __restrict__ X, const float* __restrict__ W,
                   float* __restrict__ h) {
  __shared__ float xs[16 * 260];
  const int m0  = blockIdx.x * 16;
  const int tid = threadIdx.x;

  // Cooperative, coalesced load of X tile [16][256] into padded LDS.
#pragma unroll
  for (int it = 0; it < 4; ++it) {
    const int f   = tid * 4 + it * 1024;   // flat index into 16x256 tile
    const int row = f >> 8;
    const int col = f & 255;
    const float4 v = *(const float4*)(X + (size_t)(m0 + row) * IN_F + col);
    *(float4*)(&xs[row * 260 + col]) = v;
  }
  __syncthreads();

  const int wave  = tid >> 5;
  const int lane  = tid & 31;
  const int col0  = wave * 16;
  const int l15   = lane & 15;
  const int khalf = (lane >> 4) * 2;   // lanes 0-15 -> K{0,1}; 16-31 -> K{2,3}

  // A fragment source: row l15 of LDS tile. B fragment source: W row col0+l15.
  const float* xrow = &xs[l15 * 260 + khalf];
  const float* wrow = W + (size_t)(col0 + l15) * IN_F + khalf;

  v8f acc = {};
#pragma unroll 8
  for (int k0 = 0; k0 < IN_F; k0 += 4) {
    v2f a = *(const v2f*)(xrow + k0);   // X[m0+l15][k0+khalf .. +1]
    v2f b = *(const v2f*)(wrow + k0);   // W[col0+l15][k0+khalf .. +1]
    acc = __builtin_amdgcn_wmma_f32_16x16x4_f32(
        /*neg_a=*/false, a, /*neg_b=*/false, b,
        /*c_mod=*/(short)0, acc, /*reuse_a=*/false, /*reuse_b=*/false);
  }

  // C/D layout: VGPR r -> row m0 + r (+8 for lanes 16-31), col col0 + (lane&15).
  const int mrow = m0 + ((lane >> 4) << 3);
  float* hp = h + (size_t)mrow * OUT_F + col0 + l15;
#pragma unroll
  for (int r = 0; r < 8; ++r) {
    hp[(size_t)r * OUT_F] = acc[r];
  }
}

// ---------------------------------------------------------------------------
// Kernel 2: e_src = h@att_src, e_dst = h@att_dst (wave32 per row),
// plus init denom = 1e-12 and out = 0.
// ---------------------------------------------------------------------------
__global__ __launch_bounds__(256)
void gat_rowdot_init(const float* __restrict__ h,
                     const float* __restrict__ att_src,
                     const float* __restrict__ att_dst,
                     float* __restrict__ e_src, float* __restrict__ e_dst,
                     float* __restrict__ denom, float* __restrict__ out) {
  const int wave = threadIdx.x >> 5;
  const int lane = threadIdx.x & 31;
  const int row  = blockIdx.x * 8 + wave;
  if (row >= N_NODES) return;

  const float4 hv = ((const float4*)h)[(size_t)row * 32 + lane];
  const float4 as = ((const float4*)att_src)[lane];
  const float4 ad = ((const float4*)att_dst)[lane];
  float s = hv.x * as.x + hv.y * as.y + hv.z * as.z + hv.w * as.w;
  float d = hv.x * ad.x + hv.y * ad.y + hv.z * ad.z + hv.w * ad.w;
#pragma unroll
  for (int off = 16; off > 0; off >>= 1) {
    s += __shfl_xor(s, off, 32);
    d += __shfl_xor(d, off, 32);
  }

  ((float4*)out)[(size_t)row * 32 + lane] = make_float4(0.f, 0.f, 0.f, 0.f);
  if (lane == 0) {
    e_src[row] = s;
    e_dst[row] = d;
    denom[row] = 1e-12f;
  }
}

// ---------------------------------------------------------------------------
// Kernel 3: per-edge logits. exp_e[t] = exp(clip(leakyrelu(e_src[i]+e_dst[j])))
// and denom[i] += exp_e[t] via hardware f32 atomic.
// ---------------------------------------------------------------------------
__global__ __launch_bounds__(256)
void gat_edge_logits(const int* __restrict__ ei,
                     const float* __restrict__ e_src,
                     const float* __restrict__ e_dst,
                     float* __restrict__ exp_e, float* __restrict__ denom) {
  const int t = blockIdx.x * 256 + threadIdx.x;
  if (t >= E_EDGES) return;
  const int i = ei[t];
  const int j = ei[E_EDGES + t];
  float e = e_src[i] + e_dst[j];
  e = (e >= 0.f) ? e : NEG_SLOPE * e;
  e = fminf(10.f, fmaxf(-10.f, e));
  const float ex = __expf(e);
  exp_e[t] = ex;
  unsafeAtomicAdd(&denom[i], ex);
}

// ---------------------------------------------------------------------------
// Kernel 4: wave32 per edge. Each lane covers 4 of the 128 output features:
// out[i][:] += (exp_e[t]/denom[i]) * h[j][:], via global_atomic_add_f32.
// h (51 MB) is L2-resident, so the random j-gather stays on-chip.
// ---------------------------------------------------------------------------
__global__ __launch_bounds__(256)
void gat_scatter(const int* __restrict__ ei, const float* __restrict__ exp_e,
                 const float* __restrict__ denom, const float* __restrict__ h,
                 float* __restrict__ out) {
  const int wave = threadIdx.x >> 5;
  const int lane = threadIdx.x & 31;
  const int t    = blockIdx.x * 8 + wave;
  if (t >= E_EDGES) return;
  const int i = ei[t];
  const int j = ei[E_EDGES + t];
  const float alpha = exp_e[t] / denom[i];
  const float4 hv = ((const float4*)h)[(size_t)j * 32 + lane];
  float* op = out + (size_t)i * OUT_F + lane * 4;
  unsafeAtomicAdd(op + 0, alpha * hv.x);
  unsafeAtomicAdd(op + 1, alpha * hv.y);
  unsafeAtomicAdd(op + 2, alpha * hv.z);
  unsafeAtomicAdd(op + 3, alpha * hv.w);
}

// ---------------------------------------------------------------------------
// Launch. Inputs (setup_inputs order): X, edge_index, W, att_src, att_dst.
// Workspace: h[N*128] | e_src[N] | e_dst[N] | denom[N] | exp_e[E]  (~56.4 MB).
// ---------------------------------------------------------------------------
extern "C" void kernel_launch(void* const* d_in, const int* in_sizes, int n_in,
                              void* d_out, int out_size, void* d_ws, size_t ws_size,
                              hipStream_t stream) {
  const float* X       = (const float*)d_in[0];
  const int*   ei      = (const int*)d_in[1];
  const float* W       = (const float*)d_in[2];
  const float* att_src = (const float*)d_in[3];
  const float* att_dst = (const float*)d_in[4];
  float* out = (float*)d_out;

  float* h     = (float*)d_ws;                       // N*128 floats
  float* e_src = h + (size_t)N_NODES * OUT_F;        // N
  float* e_dst = e_src + N_NODES;                    // N
  float* denom = e_dst + N_NODES;                    // N
  float* exp_e = denom + N_NODES;                    // E

  gat_gemm_wmma   <<<N_NODES / 16,        256, 0, stream>>>(X, W, h);
  gat_rowdot_init <<<N_NODES / 8,         256, 0, stream>>>(h, att_src, att_dst,
                                                            e_src, e_dst, denom, out);
  gat_edge_logits <<<(E_EDGES + 255)/256, 256, 0, stream>>>(ei, e_src, e_dst,
                                                            exp_e, denom);
  gat_scatter     <<<(E_EDGES + 7)/8,     256, 0, stream>>>(ei, exp_e, denom, h, out);
}